// VisibilityHeatmap_15599321219641
// MI455X (gfx1250) — compile-verified
//
#include <hip/hip_runtime.h>
#include <hip/hip_bf16.h>

// VisibilityHeatmap: per-keypoint heatmap gather + threshold, mask broadcast
// over the coordinate dim. Pure latency-bound gather; no matrix content, so
// no WMMA applies. Optimized for MI455X (gfx1250, wave32):
//  - one thread per (b,k) keypoint; 64-bit coalesced coord load / out store
//  - single scattered non-temporal b32 gather from the 512 MB heatmap
//  - shortest possible dependent chain per thread

#define HM_W 256
#define HM_H 256
#define HM_THRESHOLD 0.4f

// Native clang vector: accepted by __builtin_nontemporal_{load,store},
// unlike HIP's float2 class type.
typedef float v2f __attribute__((ext_vector_type(2)));

__global__ void __launch_bounds__(256)
visibility_heatmap_kernel(const float* __restrict__ coords,   // [B,K,2]
                          const float* __restrict__ heatmaps, // [B,K,H,W]
                          float* __restrict__ out,            // [B,K,2]
                          int n)                              // n = B*K
{
    int i = blockIdx.x * blockDim.x + threadIdx.x;
    if (i >= n) return;

    // Coalesced 64-bit load of (u,v) for this keypoint.
    v2f c = ((const v2f*)coords)[i];

    // NDC -> pixel, truncate toward zero (matches jnp .astype(int32)).
    float px = (c.x + 1.0f) * 0.5f * (float)HM_W;   // u -> column
    float py = (c.y + 1.0f) * 0.5f * (float)HM_H;   // v -> row
    int u = (int)px;
    int v = (int)py;

    // Validity on the unclamped indices.
    bool valid = (u > -1) & (v > -1) & (v < HM_H) & (u < HM_W);

    // Clamp for the gather (reference clips before indexing).
    int uc = min(max(u, 0), HM_W - 1);
    int vc = min(max(v, 0), HM_H - 1);

    // heatmaps[b,k,vc,uc] flat index: i*H*W + vc*W + uc (i == b*K + k).
    size_t idx = (size_t)i * (size_t)(HM_H * HM_W) + (size_t)(vc * HM_W + uc);

    // Non-temporal: the heatmap is touched exactly once, no reuse anywhere.
    float val = __builtin_nontemporal_load(heatmaps + idx);

    float m = (valid && (val > HM_THRESHOLD)) ? 1.0f : 0.0f;

    // Mask broadcast across the coordinate dim -> one b64 store.
    v2f o; o.x = m; o.y = m;
    __builtin_nontemporal_store(o, (v2f*)out + i);
}

extern "C" void kernel_launch(void* const* d_in, const int* in_sizes, int n_in,
                              void* d_out, int out_size, void* d_ws, size_t ws_size,
                              hipStream_t stream) {
    const float* coords   = (const float*)d_in[0];  // [B,K,2] float32
    const float* heatmaps = (const float*)d_in[1];  // [B,K,H,W] float32
    float* out = (float*)d_out;                     // [B,K,2] float32

    int n = in_sizes[0] / 2;                        // B*K keypoints
    int threads = 256;                              // 8 wave32s per block
    int blocks = (n + threads - 1) / threads;       // 8 blocks for B*K=2048
    visibility_heatmap_kernel<<<blocks, threads, 0, stream>>>(coords, heatmaps, out, n);
}